// Model_67851893342926
// MI455X (gfx1250) — compile-verified
//
#include <hip/hip_runtime.h>
#include <math.h>

// CDNA5 / gfx1250, wave32. WMMA f32 16x16x4 used for the clustering GEMM
// fused with a row logsumexp (softmax denominator per cell).

typedef __attribute__((ext_vector_type(2))) float v2f;
typedef __attribute__((ext_vector_type(8))) float v8f;

#define MIN_BIN_WIDTH  1e-3f
#define MIN_BIN_HEIGHT 1e-3f

// ---------------------------------------------------------------- reductions
static __device__ __forceinline__ float blockReduce256(float v, float* sm) {
  int t = threadIdx.x;
  sm[t] = v;
  __syncthreads();
#pragma unroll
  for (int st = 128; st > 0; st >>= 1) {
    if (t < st) sm[t] += sm[t + st];
    __syncthreads();
  }
  return sm[0];
}

// ---------------------------------------------------------------- K1: cluster id per cell (argmax of one-hot row)
__global__ void k_cluster(const float* __restrict__ clus, int* __restrict__ cid,
                          int C, int L) {
  int c = blockIdx.x * blockDim.x + threadIdx.x;
  if (c >= C) return;
  const float* r = clus + (size_t)c * L;
  float best = r[0];
  int bi = 0;
  for (int l = 1; l < L; ++l) {
    float v = r[l];
    if (v > best) { best = v; bi = l; }
  }
  cid[c] = bi;
}

// ---------------------------------------------------------------- K2: WMMA GEMM fused with row sum-of-exp
// bd = clustering(C x 16) @ dbw^T(16 x NT); S[c] = sum_t exp(log(bias[t]) + bd[c,t])
// negLse[c] = -log(S[c]).
// V_WMMA_F32_16X16X4_F32 operand layout (ISA 7.12.2, 32-bit 16x4 A):
//   A: lane m (0..15): v0=A[m,K=0], v1=A[m,K=1]; lane 16+m: v0=A[m,K=2], v1=A[m,K=3]
//   B (assumed mirror): lane n: v0=B[K=0,n], v1=B[K=1,n]; lane 16+n: v0=B[K=2,n], v1=B[K=3,n]
//   C/D: lane n, vgpr r -> D[M=r, N=n]; lane 16+n, vgpr r -> D[M=8+r, N=n]
__global__ void k_lse_wmma(const float* __restrict__ clus, const float* __restrict__ dbw,
                           const float* __restrict__ bias, float* __restrict__ negLse,
                           int C, int NT) {
  const int L = 16;
  int lane = threadIdx.x;
  int m = lane & 15;
  int kh = lane >> 4;  // 0: K pair {0,1}(+4ks); 1: K pair {2,3}(+4ks)
  int c0 = blockIdx.x * 16;

  // A fragments for the 4 k-steps (K=16 split as 4 x 4)
  v2f a[4];
  int row = c0 + m;
#pragma unroll
  for (int ks = 0; ks < 4; ++ks) {
    float x0 = 0.f, x1 = 0.f;
    if (row < C) {
      const float* p = clus + (size_t)row * L + ks * 4 + kh * 2;
      x0 = p[0]; x1 = p[1];
    }
    a[ks].x = x0; a[ks].y = x1;
  }

  float acc[8];
#pragma unroll
  for (int r = 0; r < 8; ++r) acc[r] = 0.f;

  int ntiles = NT >> 4;  // NT assumed multiple of 16 (10000 -> 625)
  for (int tt = 0; tt < ntiles; ++tt) {
    int tcol = tt * 16 + m;
    const float* bp = dbw + (size_t)tcol * L + kh * 2;
    v8f d = {0.f, 0.f, 0.f, 0.f, 0.f, 0.f, 0.f, 0.f};
#pragma unroll
    for (int ks = 0; ks < 4; ++ks) {
      v2f b = *(const v2f*)(bp + ks * 4);  // 8B aligned: even float offset
      d = __builtin_amdgcn_wmma_f32_16x16x4_f32(
          false, a[ks], false, b, (short)0, d, false, false);
    }
    float lb = logf(bias[tcol]);
#pragma unroll
    for (int r = 0; r < 8; ++r) acc[r] += expf(lb + d[r]);
  }

  // Row sums: reduce each 16-lane half (columns) -> 16 row sums per tile.
#pragma unroll
  for (int r = 0; r < 8; ++r) {
    float v = acc[r];
    v += __shfl_xor(v, 1, 32);
    v += __shfl_xor(v, 2, 32);
    v += __shfl_xor(v, 4, 32);
    v += __shfl_xor(v, 8, 32);
    acc[r] = v;
  }
  if (m == 0) {
#pragma unroll
    for (int r = 0; r < 8; ++r) {
      int rr = c0 + kh * 8 + r;
      if (rr < C) negLse[rr] = -logf(acc[r]);
    }
  }
}

// Fallback (only if L != 16 or NT % 16 != 0)
__global__ void k_lse_scalar(const float* __restrict__ clus, const float* __restrict__ dbw,
                             const float* __restrict__ bias, float* __restrict__ negLse,
                             int C, int NT, int L) {
  int c = blockIdx.x * blockDim.x + threadIdx.x;
  if (c >= C) return;
  const float* cr = clus + (size_t)c * L;
  float s = 0.f;
  for (int t = 0; t < NT; ++t) {
    const float* dr = dbw + (size_t)t * L;
    float bd = 0.f;
    for (int l = 0; l < L; ++l) bd += cr[l] * dr[l];
    s += expf(logf(bias[t]) + bd);
  }
  negLse[c] = -logf(s);
}

// ---------------------------------------------------------------- K3: per-gene width softmax + bin locations
__global__ void k_genewidths(const float* __restrict__ unw, const int* __restrict__ goi,
                             float* __restrict__ gW, float* __restrict__ gLoc,
                             int G, int Wt, int Ht) {
  int tid = blockIdx.x * blockDim.x + threadIdx.x;
  if (tid >= G * 3) return;
  const int nbA[3]  = {127, 63, 31};
  const int wOff[3] = {0, 127, 190};
  const int lOff[3] = {0, 128, 192};
  int g = tid / 3, s = tid % 3;
  int nb = nbA[s];
  int t = goi[g];
  const float* uw = unw + (size_t)t * Wt + wOff[s];
  float mx = uw[0];
  for (int i = 1; i < nb; ++i) mx = fmaxf(mx, uw[i]);
  float sum = 0.f;
  for (int i = 0; i < nb; ++i) sum += expf(uw[i] - mx);
  float inv = 1.f / sum;
  float scale = 1.f - MIN_BIN_WIDTH * (float)nb;
  float* wout = gW + (size_t)g * Wt + wOff[s];
  float* lout = gLoc + (size_t)g * Ht + lOff[s];
  lout[0] = 0.f;
  float run = 0.f;
  for (int i = 0; i < nb; ++i) {
    float w = MIN_BIN_WIDTH + scale * expf(uw[i] - mx) * inv;
    wout[i] = w;
    run += w;
    lout[i + 1] = (i == nb - 1) ? 1.f : run;
  }
}

// ---------------------------------------------------------------- K4: per-(cluster,gene) height / left-CDF tables
__global__ void k_comboheights(const float* __restrict__ unh, const float* __restrict__ dhw,
                               const int* __restrict__ goi, const float* __restrict__ gW,
                               float* __restrict__ cH, float* __restrict__ cCdf,
                               int G, int L, int Wt, int Ht) {
  int tid = blockIdx.x * blockDim.x + threadIdx.x;
  if (tid >= L * G * 3) return;
  const int nbA[3]  = {127, 63, 31};
  const int wOff[3] = {0, 127, 190};
  const int hOff[3] = {0, 128, 192};
  int cb = tid / 3, s = tid % 3;
  int l = cb / G, g = cb % G;  // cb = l*G + g
  int nb = nbA[s];
  int t = goi[g];
  const float* uh = unh + (size_t)t * Ht + hOff[s];
  const float* dr = dhw + ((size_t)t * L + l) * Ht + hOff[s];
  const float* w  = gW + (size_t)g * Wt + wOff[s];
  float prev = expf(uh[0] + dr[0]);
  float area = 0.f;
  for (int i = 0; i < nb; ++i) {
    float cur = expf(uh[i + 1] + dr[i + 1]);
    area += 0.5f * (prev + cur) * w[i];
    prev = cur;
  }
  float inv = 1.f / area;
  float* hout = cH + (size_t)cb * Ht + hOff[s];
  float* cout = cCdf + (size_t)cb * Ht + hOff[s];
  float hl = MIN_BIN_HEIGHT + (1.f - MIN_BIN_HEIGHT) * expf(uh[0] + dr[0]) * inv;
  hout[0] = hl;
  cout[0] = 0.f;
  float run = 0.f;
  for (int i = 0; i < nb; ++i) {
    float hr = MIN_BIN_HEIGHT + (1.f - MIN_BIN_HEIGHT) * expf(uh[i + 1] + dr[i + 1]) * inv;
    hout[i + 1] = hr;
    run += 0.5f * (hl + hr) * w[i];
    cout[i + 1] = (i == nb - 1) ? 1.f : run;
    hl = hr;
  }
}

// ---------------------------------------------------------------- K5: per-point spline chain + region term
__global__ void k_points(const float* __restrict__ coords, const int* __restrict__ lgi,
                         const int* __restrict__ lcxi, const int* __restrict__ goi,
                         const int* __restrict__ cid, const float* __restrict__ negLse,
                         const float* __restrict__ clus, const float* __restrict__ dbw,
                         const float* __restrict__ bias,
                         const float* __restrict__ gW, const float* __restrict__ gLoc,
                         const float* __restrict__ cH, const float* __restrict__ cCdf,
                         float* __restrict__ part,
                         int N, int G, int L, int Wt, int Ht, float logNT) {
  __shared__ float sm[256];
  const int nbA[3]  = {127, 63, 31};
  const int wOff[3] = {0, 127, 190};
  const int hOff[3] = {0, 128, 192};
  int n = blockIdx.x * blockDim.x + threadIdx.x;
  float v = 0.f;
  if (n < N) {
    float x = coords[n];
    int g = lgi[n];
    int cell = lcxi[n] / G;  // local_cellxgene_ix = cell*G + g
    int t = goi[g];
    int cl = cid[cell];
    size_t cb = (size_t)cl * G + g;
    float lad = 0.f;
#pragma unroll
    for (int s = 0; s < 3; ++s) {
      int nb = nbA[s];
      const float* locs = gLoc + (size_t)g * Ht + hOff[s];
      // largest i with locs[i] <= x (locs[0]=0), clipped to nb-1
      int lo = 0, hi = nb;
      while (hi - lo > 1) {
        int mid = (lo + hi) >> 1;
        if (x >= locs[mid]) lo = mid; else hi = mid;
      }
      int idx = lo < nb - 1 ? lo : nb - 1;
      float loc = locs[idx];
      float w = gW[(size_t)g * Wt + wOff[s] + idx];
      const float* H = cH + cb * Ht + hOff[s];
      float hl = H[idx], hr = H[idx + 1];
      float cdf = cCdf[cb * Ht + hOff[s] + idx];
      float alpha = (x - loc) / w;
      float out = (0.5f * (hr - hl) * alpha + hl) * w * alpha + cdf;
      out = fminf(fmaxf(out, 0.f), 1.f);
      lad += logf(alpha * (hr - hl) + hl);
      x = out;
    }
    // region likelihood: log softmax value = log(bias[t]) + bd - lse[cell]
    const float* cr = clus + (size_t)cell * L;
    const float* dr = dbw + (size_t)t * L;
    float bd = 0.f;
    for (int l = 0; l < L; ++l) bd += cr[l] * dr[l];
    v = lad + logf(bias[t]) + bd + negLse[cell] + logNT;
  }
  float bs = blockReduce256(v, sm);
  if (threadIdx.x == 0) part[blockIdx.x] = bs;
}

// ---------------------------------------------------------------- K6: KL squared-sum partials
__global__ void k_kl(const float* __restrict__ dhw, const float* __restrict__ dbw,
                     const int* __restrict__ goi, float* __restrict__ part,
                     int G, int L, int K, int total) {
  __shared__ float sm[256];
  int idx = blockIdx.x * blockDim.x + threadIdx.x;
  float v = 0.f;
  int nmix = G * L * K;
  if (idx < nmix) {
    int g = idx / (L * K);
    int off = idx - g * (L * K);
    float x = dhw[(size_t)goi[g] * L * K + off];
    v = x * x;
  } else if (idx < total) {
    int j = idx - nmix;
    int g = j / L, l = j - g * L;
    float x = dbw[(size_t)goi[g] * L + l];
    v = x * x;
  }
  float bs = blockReduce256(v, sm);
  if (threadIdx.x == 0) part[blockIdx.x] = bs;
}

// ---------------------------------------------------------------- K7: final deterministic reduce
__global__ void k_final(const float* __restrict__ pPart, int nP,
                        const float* __restrict__ kPart, int nK,
                        float* __restrict__ out, float constTerm) {
  __shared__ float sm[256];
  float a = 0.f;
  for (int i = threadIdx.x; i < nP; i += 256) a += pPart[i];
  float sP = blockReduce256(a, sm);
  __syncthreads();
  float b = 0.f;
  for (int i = threadIdx.x; i < nK; i += 256) b += kPart[i];
  float sK = blockReduce256(b, sm);
  if (threadIdx.x == 0) {
    // result = elbo - mixture_kl.sum() - baseline_kl.sum()
    //        = -(sum of lad+region) + 0.5*sum(x^2) + count*0.5*log(2*pi)
    out[0] = -sP + 0.5f * sK + constTerm;
  }
}

// ---------------------------------------------------------------- launcher
extern "C" void kernel_launch(void* const* d_in, const int* in_sizes, int n_in,
                              void* d_out, int out_size, void* d_ws, size_t ws_size,
                              hipStream_t stream) {
  const float* coords = (const float*)d_in[0];
  const float* clus   = (const float*)d_in[1];
  const float* dhw    = (const float*)d_in[2];   // (NT, L, K)
  const float* dbw    = (const float*)d_in[3];   // (NT, L)
  const float* unh    = (const float*)d_in[4];   // (NT, K)
  const float* unw    = (const float*)d_in[5];   // (NT, W)
  const float* bias   = (const float*)d_in[6];   // (NT,)
  const int*   goi    = (const int*)d_in[7];     // (G,)
  const int*   lgi    = (const int*)d_in[8];     // (N,)
  const int*   lcxi   = (const int*)d_in[9];     // (N,) cell*G + g
  // d_in[10] localcellxgene_ix: redundant given lcxi/goi

  int N  = in_sizes[0];
  int NT = in_sizes[6];
  int L  = in_sizes[3] / NT;   // 16
  int C  = in_sizes[1] / L;    // 1000
  int K  = in_sizes[4] / NT;   // 224 heights total
  int W  = in_sizes[5] / NT;   // 221 widths total
  int G  = in_sizes[7];        // 500

  // carve workspace (~15.3 MB), 256B aligned slots
  char* p = (char*)d_ws;
  auto carve = [&](size_t bytes) -> char* {
    char* r = p;
    p += (bytes + 255) & ~(size_t)255;
    return r;
  };
  int*   cid    = (int*)  carve((size_t)C * 4);
  float* negLse = (float*)carve((size_t)C * 4);
  float* gW     = (float*)carve((size_t)G * W * 4);
  float* gLoc   = (float*)carve((size_t)G * K * 4);
  float* cH     = (float*)carve((size_t)L * G * K * 4);
  float* cCdf   = (float*)carve((size_t)L * G * K * 4);
  int nP = (N + 255) / 256;
  float* pPart  = (float*)carve((size_t)nP * 4);
  int total = G * L * K + G * L;
  int nK = (total + 255) / 256;
  float* kPart  = (float*)carve((size_t)nK * 4);
  (void)ws_size; (void)n_in; (void)out_size;

  k_cluster<<<(C + 255) / 256, 256, 0, stream>>>(clus, cid, C, L);

  if (L == 16 && (NT & 15) == 0) {
    k_lse_wmma<<<(C + 15) / 16, 32, 0, stream>>>(clus, dbw, bias, negLse, C, NT);
  } else {
    k_lse_scalar<<<(C + 255) / 256, 256, 0, stream>>>(clus, dbw, bias, negLse, C, NT, L);
  }

  k_genewidths<<<(G * 3 + 127) / 128, 128, 0, stream>>>(unw, goi, gW, gLoc, G, W, K);
  k_comboheights<<<(L * G * 3 + 127) / 128, 128, 0, stream>>>(unh, dhw, goi, gW, cH, cCdf,
                                                              G, L, W, K);

  float logNT = logf((float)NT);
  k_points<<<nP, 256, 0, stream>>>(coords, lgi, lcxi, goi, cid, negLse, clus, dbw, bias,
                                   gW, gLoc, cH, cCdf, pPart, N, G, L, W, K, logNT);

  k_kl<<<nK, 256, 0, stream>>>(dhw, dbw, goi, kPart, G, L, K, total);

  const float HL2PI = 0.91893853320467274178f;  // 0.5*log(2*pi)
  float constTerm = HL2PI * (float)((long long)G * L * K + (long long)G * L);
  k_final<<<1, 256, 0, stream>>>(pPart, nP, kPart, nK, (float*)d_out, constTerm);
}